// BatchedKiloNeRF_1039382086436
// MI455X (gfx1250) — compile-verified
//
#include <hip/hip_runtime.h>

#define NUM_MODELS 4096
#define W 32
#define N_PTS 131072

typedef __attribute__((ext_vector_type(2))) float v2f;
typedef __attribute__((ext_vector_type(8))) float v8f;

// D(16x16,f32) = A(16x4,f32) x B(4x16,f32) + C
__device__ __forceinline__ v8f wmma4(v2f a, v2f b, v8f c) {
    return __builtin_amdgcn_wmma_f32_16x16x4_f32(
        /*neg_a=*/false, a, /*neg_b=*/false, b,
        /*c_mod=*/(short)0, c, /*reuse_a=*/false, /*reuse_b=*/false);
}

// ---------------- Phase A: bucket points by model ----------------

__global__ void kn_zero_counts(int* __restrict__ counts) {
    counts[blockIdx.x * 256 + threadIdx.x] = 0;
}

__global__ void kn_count(const int* __restrict__ mi, int* __restrict__ counts) {
    int i = blockIdx.x * 256 + threadIdx.x;
    if (i < N_PTS) atomicAdd(&counts[mi[i]], 1);
}

// single block, 1024 threads, 4 models each
__global__ void kn_scan(const int* __restrict__ counts,
                        int* __restrict__ offsets, int* __restrict__ cursor) {
    __shared__ int partial[1024];
    int t = threadIdx.x;
    int local[4];
    int s = 0;
    #pragma unroll
    for (int j = 0; j < 4; ++j) { local[j] = counts[t * 4 + j]; s += local[j]; }
    partial[t] = s;
    __syncthreads();
    for (int off = 1; off < 1024; off <<= 1) {
        int v = (t >= off) ? partial[t - off] : 0;
        __syncthreads();
        partial[t] += v;
        __syncthreads();
    }
    int excl = (t == 0) ? 0 : partial[t - 1];
    #pragma unroll
    for (int j = 0; j < 4; ++j) {
        offsets[t * 4 + j] = excl;
        cursor[t * 4 + j]  = excl;
        excl += local[j];
    }
}

__global__ void kn_scatter(const int* __restrict__ mi,
                           int* __restrict__ cursor, int* __restrict__ order) {
    int i = blockIdx.x * 256 + threadIdx.x;
    if (i < N_PTS) {
        int pos = atomicAdd(&cursor[mi[i]], 1);
        order[pos] = i;
    }
}

// ---------------- Phase B: per-model MLP with f32 WMMA ----------------

__launch_bounds__(32)
__global__ void kn_mlp(
    const float* __restrict__ x, const int* __restrict__ order,
    const int* __restrict__ offsets, const int* __restrict__ counts,
    const float* __restrict__ pts_w0, const float* __restrict__ pts_b0,
    const float* __restrict__ pts_w1, const float* __restrict__ pts_b1,
    const float* __restrict__ feat_w, const float* __restrict__ feat_b,
    const float* __restrict__ sigma_w, const float* __restrict__ sigma_b,
    const float* __restrict__ view_w, const float* __restrict__ view_b,
    const float* __restrict__ rgb_w, const float* __restrict__ rgb_b,
    float* __restrict__ out)
{
    const int m = blockIdx.x;
    const int cnt = counts[m];
    if (cnt == 0) return;
    const int base = offsets[m];
    const int lane = threadIdx.x;
    const int lo = lane & 15;
    const int hi = lane >> 4;

    __shared__ __attribute__((aligned(16))) float hA[16][4];   // layer0 input (pts, padded)
    __shared__ __attribute__((aligned(16))) float vt[16][4];   // views (padded)
    __shared__ __attribute__((aligned(16))) float t0[16][36];  // activation tile A
    __shared__ __attribute__((aligned(16))) float t1[16][36];  // activation tile B
    __shared__ float sw_s[32];   // sigma weights
    __shared__ float rw_s[96];   // rgb weights (3x32)
    __shared__ float rb_s[4];    // rgb_b[0..2], sigma_b in [3]

    // small head weights -> LDS
    sw_s[lane] = sigma_w[m * 32 + lane];
    for (int j = lane; j < 96; j += 32) rw_s[j] = rgb_w[m * 96 + j];
    if (lane < 3) rb_s[lane] = rgb_b[m * 3 + lane];
    if (lane == 3) rb_s[3] = sigma_b[m];

    // ---- Load all B-matrix fragments once (B = W^T, ISA 4x16 layout) ----
    // lane holds N = nb*16+lo ; VGPR pair covers K = kc*4 + 2*hi, +1
    const float* w0p = pts_w0 + (size_t)m * W * 3;
    const float* w1p = pts_w1 + (size_t)m * W * W;
    const float* fwp = feat_w + (size_t)m * W * W;
    const float* vwp = view_w + (size_t)m * W * (W + 3);

    v2f Bw0[2], Bw1[2][8], Bf[2][8], Bv[2][9];
    #pragma unroll
    for (int nb = 0; nb < 2; ++nb) {
        const int n = nb * 16 + lo;
        {   // pts_w0: 32x3, K padded to 4 with zeros
            int kb = 2 * hi;
            float bx = (kb < 3) ? w0p[n * 3 + kb] : 0.f;
            float by = (kb + 1 < 3) ? w0p[n * 3 + kb + 1] : 0.f;
            Bw0[nb] = (v2f){bx, by};
        }
        #pragma unroll
        for (int kc = 0; kc < 8; ++kc) {
            int kb = kc * 4 + 2 * hi;
            Bw1[nb][kc] = (v2f){ w1p[n * W + kb], w1p[n * W + kb + 1] };
            Bf[nb][kc]  = (v2f){ fwp[n * W + kb], fwp[n * W + kb + 1] };
            Bv[nb][kc]  = (v2f){ vwp[n * 35 + kb], vwp[n * 35 + kb + 1] };
        }
        {   // view inputs 32..34 (K chunk 8, padded to 36)
            int kb = 32 + 2 * hi;
            float bx = (kb < 35) ? vwp[n * 35 + kb] : 0.f;
            float by = (kb + 1 < 35) ? vwp[n * 35 + kb + 1] : 0.f;
            Bv[nb][8] = (v2f){bx, by};
        }
    }
    // biases: D column = nb*16 + lo, constant per lane
    const float b0c[2] = { pts_b0[m * W + lo], pts_b0[m * W + 16 + lo] };
    const float b1c[2] = { pts_b1[m * W + lo], pts_b1[m * W + 16 + lo] };
    const float bfc[2] = { feat_b[m * W + lo], feat_b[m * W + 16 + lo] };
    const float bvc[2] = { view_b[m * W + lo], view_b[m * W + 16 + lo] };

    const int ntiles = (cnt + 15) >> 4;
    for (int tile = 0; tile < ntiles; ++tile) {
        const int tbase = base + tile * 16;
        int valid = cnt - tile * 16; if (valid > 16) valid = 16;
        int pt = -1;

        __syncthreads();
        if (lane < 16) {
            float xv[6] = {0.f, 0.f, 0.f, 0.f, 0.f, 0.f};
            if (lane < valid) {
                pt = order[tbase + lane];
                const float* xp = x + (size_t)pt * 6;
                #pragma unroll
                for (int j = 0; j < 6; ++j) xv[j] = xp[j];
            }
            hA[lane][0] = xv[0]; hA[lane][1] = xv[1]; hA[lane][2] = xv[2]; hA[lane][3] = 0.f;
            vt[lane][0] = xv[3]; vt[lane][1] = xv[4]; vt[lane][2] = xv[5]; vt[lane][3] = 0.f;
        }
        __syncthreads();

        // ---- layer0: h0 = relu(x3 @ W0^T + b0) ----
        v2f a0 = *(const v2f*)&hA[lo][2 * hi];
        v8f z = {};
        v8f d0 = wmma4(a0, Bw0[0], z);
        v8f d1 = wmma4(a0, Bw0[1], z);
        #pragma unroll
        for (int r = 0; r < 8; ++r) {
            int row = r + 8 * hi;
            t0[row][lo]      = fmaxf(d0[r] + b0c[0], 0.f);
            t0[row][16 + lo] = fmaxf(d1[r] + b0c[1], 0.f);
        }
        __syncthreads();

        // ---- layer1: h1 = relu(h0 @ W1^T + b1) ----
        v8f e0 = {}, e1 = {};
        #pragma unroll
        for (int kc = 0; kc < 8; ++kc) {
            v2f a = *(const v2f*)&t0[lo][kc * 4 + 2 * hi];
            e0 = wmma4(a, Bw1[0][kc], e0);
            e1 = wmma4(a, Bw1[1][kc], e1);
        }
        __syncthreads();
        #pragma unroll
        for (int r = 0; r < 8; ++r) {
            int row = r + 8 * hi;
            t1[row][lo]      = fmaxf(e0[r] + b1c[0], 0.f);
            t1[row][16 + lo] = fmaxf(e1[r] + b1c[1], 0.f);
        }
        __syncthreads();

        // ---- sigma head (per point, from h1 in t1) ----
        float sig = 0.f;
        if (lane < 16) {
            sig = rb_s[3];
            #pragma unroll 4
            for (int i = 0; i < 32; ++i) sig += t1[lane][i] * sw_s[i];
        }

        // ---- feature = h1 @ Wf^T + bf (no relu); append views ----
        v8f f0 = {}, f1 = {};
        #pragma unroll
        for (int kc = 0; kc < 8; ++kc) {
            v2f a = *(const v2f*)&t1[lo][kc * 4 + 2 * hi];
            f0 = wmma4(a, Bf[0][kc], f0);
            f1 = wmma4(a, Bf[1][kc], f1);
        }
        __syncthreads();
        #pragma unroll
        for (int r = 0; r < 8; ++r) {
            int row = r + 8 * hi;
            t0[row][lo]      = f0[r] + bfc[0];
            t0[row][16 + lo] = f1[r] + bfc[1];
        }
        if (lane < 16) {
            t0[lane][32] = vt[lane][0];
            t0[lane][33] = vt[lane][1];
            t0[lane][34] = vt[lane][2];
            t0[lane][35] = 0.f;
        }
        __syncthreads();

        // ---- view layer: h3 = relu([feat, views] @ Wv^T + bv), K=36 padded ----
        v8f g0 = {}, g1 = {};
        #pragma unroll
        for (int kc = 0; kc < 9; ++kc) {
            v2f a = *(const v2f*)&t0[lo][kc * 4 + 2 * hi];
            g0 = wmma4(a, Bv[0][kc], g0);
            g1 = wmma4(a, Bv[1][kc], g1);
        }
        __syncthreads();
        #pragma unroll
        for (int r = 0; r < 8; ++r) {
            int row = r + 8 * hi;
            t1[row][lo]      = fmaxf(g0[r] + bvc[0], 0.f);
            t1[row][16 + lo] = fmaxf(g1[r] + bvc[1], 0.f);
        }
        __syncthreads();

        // ---- rgb head + store out[pt] = {r,g,b,sigma} ----
        if (lane < valid) {
            float r0 = rb_s[0], r1 = rb_s[1], r2 = rb_s[2];
            #pragma unroll 4
            for (int i = 0; i < 32; ++i) {
                float hv = t1[lane][i];
                r0 += hv * rw_s[i];
                r1 += hv * rw_s[32 + i];
                r2 += hv * rw_s[64 + i];
            }
            float* op = out + (size_t)pt * 4;
            op[0] = r0; op[1] = r1; op[2] = r2; op[3] = sig;
        }
    }
}

// ---------------- host launch ----------------

extern "C" void kernel_launch(void* const* d_in, const int* in_sizes, int n_in,
                              void* d_out, int out_size, void* d_ws, size_t ws_size,
                              hipStream_t stream) {
    (void)in_sizes; (void)n_in; (void)out_size; (void)ws_size;
    const float* x        = (const float*)d_in[0];
    const int*   mi       = (const int*)d_in[1];
    const float* pts_w0   = (const float*)d_in[2];
    const float* pts_b0   = (const float*)d_in[3];
    const float* pts_w1   = (const float*)d_in[4];
    const float* pts_b1   = (const float*)d_in[5];
    const float* feat_w   = (const float*)d_in[6];
    const float* feat_b   = (const float*)d_in[7];
    const float* sigma_w  = (const float*)d_in[8];
    const float* sigma_b  = (const float*)d_in[9];
    const float* view_w   = (const float*)d_in[10];
    const float* view_b   = (const float*)d_in[11];
    const float* rgb_w    = (const float*)d_in[12];
    const float* rgb_b    = (const float*)d_in[13];
    float* out = (float*)d_out;

    int* counts  = (int*)d_ws;                    // 4096
    int* offsets = counts + NUM_MODELS;           // 4096
    int* cursor  = offsets + NUM_MODELS;          // 4096
    int* order   = cursor + NUM_MODELS;           // 131072

    kn_zero_counts<<<NUM_MODELS / 256, 256, 0, stream>>>(counts);
    kn_count<<<N_PTS / 256, 256, 0, stream>>>(mi, counts);
    kn_scan<<<1, 1024, 0, stream>>>(counts, offsets, cursor);
    kn_scatter<<<N_PTS / 256, 256, 0, stream>>>(mi, cursor, order);
    kn_mlp<<<NUM_MODELS, 32, 0, stream>>>(
        x, order, offsets, counts,
        pts_w0, pts_b0, pts_w1, pts_b1,
        feat_w, feat_b, sigma_w, sigma_b,
        view_w, view_b, rgb_w, rgb_b, out);
}